// GNNConv_32315333935196
// MI455X (gfx1250) — compile-verified
//
#include <hip/hip_runtime.h>

// GCNConv (symmetric norm, self loops) + LayerNorm, plus LN(x_ @ W), fp32.
// GEMMs use V_WMMA_F32_16X16X4_F32 (the only dense fp32 WMMA on gfx1250).

#define D 128

typedef __attribute__((ext_vector_type(2))) float v2f;
typedef __attribute__((ext_vector_type(8))) float v8f;

__device__ __forceinline__ float wave_sum(float v) {
#pragma unroll
  for (int off = 16; off > 0; off >>= 1) v += __shfl_xor(v, off, 32);
  return v;
}

// ---------------- degree / normalization ----------------

__global__ void deg_init_kernel(float* __restrict__ deg, int n) {
  int i = blockIdx.x * 256 + threadIdx.x;
  if (i < n) deg[i] = 1.0f;  // self loop contributes 1
}

__global__ void deg_count_kernel(const int* __restrict__ dst, float* __restrict__ deg, int ne) {
  int i = blockIdx.x * 256 + threadIdx.x;
  if (i < ne) unsafeAtomicAdd(&deg[dst[i]], 1.0f);
}

__global__ void deg_rsqrt_kernel(float* __restrict__ deg, int n) {
  int i = blockIdx.x * 256 + threadIdx.x;
  if (i < n) deg[i] = rsqrtf(deg[i]);  // deg >= 1 always
}

// ---------------- fp32 WMMA GEMM: Y[n,128] = X[n,128] @ W[128,128] ----------------
// Block = 256 threads = 8 waves; block handles 16 rows, wave w handles cols [16w,16w+16).
// A frag (16x4 f32, 2 VGPR): lanes 0-15 -> M rows, K=0,1; lanes 16-31 -> K=2,3 (float2 load).
// B frag (4x16 f32, 2 VGPR): lanes hold N; VGPR0/1 = K {0,1} (lo half) / {2,3} (hi half).
// C/D (16x16 f32, 8 VGPR): VGPR r = row hi*8+r, col = lane%16.

__device__ __forceinline__ v8f gemm_tile_f32(const float* __restrict__ X,
                                             const float* __restrict__ Wm,
                                             int arow, int col, int hi) {
  const float* xrow = X + (size_t)arow * D + 2 * hi;
  v8f acc = {};
#pragma unroll 4
  for (int k = 0; k < D; k += 4) {
    v2f a = *(const v2f*)(xrow + k);  // X[arow][k+2*hi .. +1], 8B aligned
    v2f b;
    b.x = Wm[(k + 2 * hi) * D + col];
    b.y = Wm[(k + 2 * hi + 1) * D + col];
    acc = __builtin_amdgcn_wmma_f32_16x16x4_f32(false, a, false, b, (short)0, acc, false, false);
  }
  return acc;
}

__global__ __launch_bounds__(256) void gemm_f32_wmma_kernel(
    const float* __restrict__ X, const float* __restrict__ Wm,
    float* __restrict__ Y, int nrows) {
  const int wave = threadIdx.x >> 5;
  const int lane = threadIdx.x & 31;
  const int lr   = lane & 15;
  const int hi   = lane >> 4;
  const int row_base = blockIdx.x * 16;
  const int col = (wave << 4) + lr;

  int arow = row_base + lr;
  if (arow > nrows - 1) arow = nrows - 1;  // clamp loads, keep EXEC all-ones for WMMA
  v8f acc = gemm_tile_f32(X, Wm, arow, col, hi);

  if (row_base + 16 <= nrows) {
    // uniform fast path: whole tile in range, branch-free stores
    float* yrow = Y + (size_t)(row_base + hi * 8) * D + col;
#pragma unroll
    for (int r = 0; r < 8; ++r) yrow[(size_t)r * D] = acc[r];
  } else {
#pragma unroll
    for (int r = 0; r < 8; ++r) {
      int row = row_base + hi * 8 + r;
      if (row < nrows) Y[(size_t)row * D + col] = acc[r];
    }
  }
}

// ---------------- fused GEMM + LayerNorm (branch 2): out2 = LN(x_ @ W) ----------------

__global__ __launch_bounds__(256) void gemm_ln_wmma_kernel(
    const float* __restrict__ X, const float* __restrict__ Wm,
    const float* __restrict__ gamma, const float* __restrict__ beta,
    float* __restrict__ Out, int nrows) {
  __shared__ float smem[16][D];
  const int wave = threadIdx.x >> 5;
  const int lane = threadIdx.x & 31;
  const int lr   = lane & 15;
  const int hi   = lane >> 4;
  const int row_base = blockIdx.x * 16;
  const int col = (wave << 4) + lr;

  int arow = row_base + lr;
  if (arow > nrows - 1) arow = nrows - 1;
  v8f acc = gemm_tile_f32(X, Wm, arow, col, hi);

#pragma unroll
  for (int r = 0; r < 8; ++r) smem[hi * 8 + r][col] = acc[r];
  __syncthreads();

  // LayerNorm: each wave normalizes rows 2*wave and 2*wave+1 (128 cols, 4 per lane)
#pragma unroll
  for (int rr = 0; rr < 2; ++rr) {
    int row = wave * 2 + rr;
    int grow = row_base + row;
    float v0 = smem[row][lane];
    float v1 = smem[row][lane + 32];
    float v2 = smem[row][lane + 64];
    float v3 = smem[row][lane + 96];
    float mu = wave_sum(v0 + v1 + v2 + v3) * (1.0f / D);
    float d0 = v0 - mu, d1 = v1 - mu, d2 = v2 - mu, d3 = v3 - mu;
    float var = wave_sum(d0 * d0 + d1 * d1 + d2 * d2 + d3 * d3) * (1.0f / D);
    float rstd = rsqrtf(var + 1e-5f);
    if (grow < nrows) {  // wave-uniform guard
      float* orow = Out + (size_t)grow * D;
      orow[lane]      = d0 * rstd * gamma[lane]      + beta[lane];
      orow[lane + 32] = d1 * rstd * gamma[lane + 32] + beta[lane + 32];
      orow[lane + 64] = d2 * rstd * gamma[lane + 64] + beta[lane + 64];
      orow[lane + 96] = d3 * rstd * gamma[lane + 96] + beta[lane + 96];
    }
  }
}

// ---------------- self-loop init: agg[i] = xt[i] * dinv[i]^2 (overwrites poisoned d_out) --------

__global__ void agg_init_kernel(const float* __restrict__ xt, const float* __restrict__ dinv,
                                float* __restrict__ agg, int n) {
  long long i = (long long)blockIdx.x * 256 + threadIdx.x;
  if (i < (long long)n * D) {
    int row = (int)(i >> 7);
    float di = dinv[row];
    agg[i] = xt[i] * (di * di);
  }
}

// ---------------- edge scatter: agg[dst] += xt[src] * dinv[src]*dinv[dst] ----------------
// One wave per edge: lane handles channels lane, +32, +64, +96 (128B coalesced rows).
// xt (51MB) and agg (51MB) both fit in the 192MB L2 -> gather+atomics run at L2 BW.

__global__ __launch_bounds__(256) void scatter_kernel(
    const float* __restrict__ xt, const float* __restrict__ dinv,
    const int* __restrict__ src, const int* __restrict__ dst,
    float* __restrict__ agg, int ne) {
  int wave = threadIdx.x >> 5;
  int lane = threadIdx.x & 31;
  int e = blockIdx.x * 8 + wave;
  if (e >= ne) return;  // wave-uniform exit
  int s = src[e], d = dst[e];
  float ew = dinv[s] * dinv[d];
  const float* xr = xt + (size_t)s * D;
  float* orow = agg + (size_t)d * D;
#pragma unroll
  for (int i = 0; i < 4; ++i)
    unsafeAtomicAdd(&orow[lane + 32 * i], xr[lane + 32 * i] * ew);
}

// ---------------- in-place LN with GCN bias (branch 1) ----------------

__global__ __launch_bounds__(256) void ln_bias_kernel(
    float* __restrict__ A, const float* __restrict__ bias,
    const float* __restrict__ gamma, const float* __restrict__ beta, int nrows) {
  int wave = threadIdx.x >> 5;
  int lane = threadIdx.x & 31;
  int row = blockIdx.x * 8 + wave;
  if (row >= nrows) return;  // wave-uniform exit
  float* arow = A + (size_t)row * D;
  float v0 = arow[lane]      + bias[lane];
  float v1 = arow[lane + 32] + bias[lane + 32];
  float v2 = arow[lane + 64] + bias[lane + 64];
  float v3 = arow[lane + 96] + bias[lane + 96];
  float mu = wave_sum(v0 + v1 + v2 + v3) * (1.0f / D);
  float d0 = v0 - mu, d1 = v1 - mu, d2 = v2 - mu, d3 = v3 - mu;
  float var = wave_sum(d0 * d0 + d1 * d1 + d2 * d2 + d3 * d3) * (1.0f / D);
  float rstd = rsqrtf(var + 1e-5f);
  arow[lane]      = d0 * rstd * gamma[lane]      + beta[lane];
  arow[lane + 32] = d1 * rstd * gamma[lane + 32] + beta[lane + 32];
  arow[lane + 64] = d2 * rstd * gamma[lane + 64] + beta[lane + 64];
  arow[lane + 96] = d3 * rstd * gamma[lane + 96] + beta[lane + 96];
}

extern "C" void kernel_launch(void* const* d_in, const int* in_sizes, int n_in,
                              void* d_out, int out_size, void* d_ws, size_t ws_size,
                              hipStream_t stream) {
  const float* x      = (const float*)d_in[0];
  const float* x_     = (const float*)d_in[1];
  const float* Wm     = (const float*)d_in[2];
  const float* b      = (const float*)d_in[3];
  const float* gamma1 = (const float*)d_in[4];
  const float* beta1  = (const float*)d_in[5];
  const float* gamma2 = (const float*)d_in[6];
  const float* beta2  = (const float*)d_in[7];
  const int*   eidx   = (const int*)d_in[8];

  const int n  = in_sizes[0] / D;   // 100000
  const int ne = in_sizes[8] / 2;   // 1600000
  const int* src = eidx;
  const int* dst = eidx + ne;

  float* out1 = (float*)d_out;                 // LN(agg + b)
  float* out2 = out1 + (size_t)n * D;          // LN(x_ @ W)

  // workspace: dinv[n] | xt[n*D]   (~51.6 MB)
  float* dinv = (float*)d_ws;
  float* xt   = dinv + ((n + 63) & ~63);

  const int tiles  = (n + 15) / 16;
  const int nthr   = (n + 255) / 256;
  const int nethr  = (ne + 255) / 256;
  const int newave = (ne + 7) / 8;
  const int nwave  = (n + 7) / 8;
  const long long elems = (long long)n * D;
  const int eblk = (int)((elems + 255) / 256);

  // 1) degree (incl. self loop) -> dinv = rsqrt(deg)
  deg_init_kernel<<<nthr, 256, 0, stream>>>(dinv, n);
  deg_count_kernel<<<nethr, 256, 0, stream>>>(dst, dinv, ne);
  deg_rsqrt_kernel<<<nthr, 256, 0, stream>>>(dinv, n);

  // 2) xt = x @ W  (fp32 WMMA)
  gemm_f32_wmma_kernel<<<tiles, 256, 0, stream>>>(x, Wm, xt, n);

  // 3) agg := self-loop term, then scatter-add edges (atomics into d_out, L2-resident)
  agg_init_kernel<<<eblk, 256, 0, stream>>>(xt, dinv, out1, n);
  scatter_kernel<<<newave, 256, 0, stream>>>(xt, dinv, src, dst, out1, ne);

  // 4) branch 1: LN(agg + b) in place
  ln_bias_kernel<<<nwave, 256, 0, stream>>>(out1, b, gamma1, beta1, n);

  // 5) branch 2: fused GEMM + LN, never materializes x2
  gemm_ln_wmma_kernel<<<tiles, 256, 0, stream>>>(x_, Wm, gamma2, beta2, out2, n);
}